// NormalizedEKVConv2d_27144193310958
// MI455X (gfx1250) — compile-verified
//
#include <hip/hip_runtime.h>
#include <hip/hip_bf16.h>
#include <stdint.h>

typedef float        v2f   __attribute__((ext_vector_type(2)));
typedef float        v8f   __attribute__((ext_vector_type(8)));
typedef unsigned int u32x4 __attribute__((ext_vector_type(4)));
typedef int          i32x4 __attribute__((ext_vector_type(4)));
typedef int          i32x8 __attribute__((ext_vector_type(8)));

#define K_N          2
#define K_C          64
#define K_H          32
#define K_W          32
#define K_COUT       64
#define K_FAN        576                   // 64 * 3 * 3
#define K_THETA_ELTS (K_COUT * K_FAN)      // 36864 floats = 144 KB
#define K_XS_ELTS    (K_C * 3 * 34)        // 6528 floats: [c][3 rows][34 padded cols]

// softplus(a*ln2)/ln2 = max(a,0) + log2(1 + 2^-|a|)  -> 1 v_exp_f32 + 1 v_log_f32
__device__ __forceinline__ float sp_h(float a) {
  float e = __builtin_amdgcn_exp2f(-__builtin_fabsf(a));
  float l = __builtin_amdgcn_logf(1.0f + e);     // hardware log2
  return __builtin_fmaxf(a, 0.0f) + l;
}

// (softplus(v/phi)^2 - softplus((v-VD)/phi)^2) / ln2^2   (alpha*ln2^2*tia folded later)
__device__ __forceinline__ float ekv_g(float v) {
  const float S  = 19.235933878519513f;   // log2(e) / (2*1.5*0.025)
  const float D2 = 9.6179669392597564f;   // 0.5 * S  (= VD/phi in log2 units)
  float a  = v * S;
  float h1 = sp_h(a);
  float h2 = sp_h(a - D2);
  return (h1 - h2) * (h1 + h2);
}

__global__ __launch_bounds__(256) void ekv_conv_kernel(
    const float* __restrict__ x, const float* __restrict__ theta,
    float* __restrict__ out)
{
  extern __shared__ float smem[];
  float* ths = smem;                    // theta[co][f], 36864 floats (TDM-staged)
  float* xs  = smem + K_THETA_ELTS;     // x slab  [c][r=3][cp=34], zero-padded

  const int tid  = threadIdx.x;
  const int lane = tid & 31;
  const int wv   = tid >> 5;
  const int n    = blockIdx.x >> 5;
  const int h    = blockIdx.x & 31;

  // ---- TDM: stage all of theta (144 KB) into LDS with one descriptor ----
  if (tid == 0) {
    unsigned long long ga = (unsigned long long)(uintptr_t)theta;
    unsigned lds = (unsigned)(uintptr_t)ths;        // flat low 32 bits == LDS offset
    const unsigned ne = K_THETA_ELTS;               // 36864 elements (fits 16-bit tile_dim0)
    u32x4 g0;
    g0[0] = 1u;                                     // count=1, user mode, no gather
    g0[1] = lds;                                    // lds_addr
    g0[2] = (unsigned)(ga & 0xffffffffull);         // global_addr[31:0]
    g0[3] = (unsigned)((ga >> 32) & 0x1ffffffull)   // global_addr[56:32]
          | (2u << 30);                             // type = 2 ("image")
    i32x8 g1;
    g1[0] = (int)(2u << 16);                        // data_size = 4 bytes
    g1[1] = (int)((ne & 0xffffu) << 16);            // tensor_dim0[15:0]
    g1[2] = (int)(((ne >> 16) & 0xffffu)            // tensor_dim0[31:16]
          | (1u << 16));                            // tensor_dim1 = 1
    g1[3] = (int)((ne & 0xffffu) << 16);            // tile_dim0 = 36864
    g1[4] = 1;                                      // tile_dim1 = 1, tile_dim2 = 0
    g1[5] = (int)ne;                                // tensor_dim0_stride[31:0]
    g1[6] = (int)((ne & 0xffffu) << 16);            // tensor_dim1_stride[15:0]
    g1[7] = 0;
    i32x4 g2 = {0, 0, 0, 0};
    i32x4 g3 = {0, 0, 0, 0};
#if defined(__clang_major__) && (__clang_major__ >= 23)
    i32x8 gz = {0, 0, 0, 0, 0, 0, 0, 0};
    __builtin_amdgcn_tensor_load_to_lds(g0, g1, g2, g3, gz, 0);
#else
    __builtin_amdgcn_tensor_load_to_lds(g0, g1, g2, g3, 0);
#endif
  }

  // ---- stage zero-padded x row slab (overlaps with the TDM transfer) ----
  for (int idx = tid; idx < K_XS_ELTS; idx += 256) {
    int c   = idx / 102;          // 102 = 3*34
    int r2  = idx - c * 102;
    int r   = r2 / 34;
    int cp  = r2 - r * 34;
    int hr  = h + r - 1;
    int col = cp - 1;
    float v = 0.0f;
    if ((unsigned)hr < 32u && (unsigned)col < 32u)
      v = x[((n * K_C + c) * K_H + hr) * K_W + col];
    xs[idx] = v;
  }

  __builtin_amdgcn_s_wait_tensorcnt(0);
  __syncthreads();

  const float SCALE = (float)(2e-6 * (500000.0 / 24.0) *
                              0.6931471805599453 * 0.6931471805599453);
  const v2f Bones = {1.0f, 1.0f};       // B = ones(4x16): exact fp32 reduction
  const int koff   = (lane >> 4) << 1;  // A layout: lanes 16-31 hold K=2..3
  const int mrow   = lane & 15;         // A layout: M = lane & 15
  const int cobase = wv << 3;           // 8 output channels per wave

  for (int pg = 0; pg < 2; ++pg) {
    const unsigned pix = (unsigned)((pg << 4) + mrow);   // output column w

    v8f zero = {0.f, 0.f, 0.f, 0.f, 0.f, 0.f, 0.f, 0.f};
    v8f acc[8];
    #pragma unroll
    for (int j = 0; j < 8; ++j) acc[j] = zero;

    for (int fb = 0; fb < K_FAN; fb += 4) {
      const unsigned uf0 = (unsigned)(fb + koff);
      const unsigned uf1 = uf0 + 1u;
      unsigned c0 = uf0 / 9u, q0 = uf0 - c0 * 9u;
      unsigned y0 = q0 / 3u,  k0 = q0 - y0 * 3u;
      unsigned c1 = uf1 / 9u, q1 = uf1 - c1 * 9u;
      unsigned y1 = q1 / 3u,  k1 = q1 - y1 * 3u;

      // --- batch ALL LDS traffic for this chunk: 2 x-reads + 16 theta reads,
      //     then one dscnt wait instead of 8 interleaved load/wait pairs ---
      float xv0 = xs[(c0 * 3u + y0) * 34u + pix + k0];   // conflict-free
      float xv1 = xs[(c1 * 3u + y1) * 34u + pix + k1];
      float w0[8], w1[8];
      #pragma unroll
      for (int j = 0; j < 8; ++j) {
        const float* wrow = ths + (cobase + j) * K_FAN;  // LDS broadcast reads
        w0[j] = wrow[uf0];
        w1[j] = wrow[uf1];
      }

      // --- 16 independent g-evals = 32 exp2 + 32 log2 in flight: deep TRANS
      //     pool lets the scheduler hide the TRANS32 hazard (no v_nop stalls) ---
      v2f Av[8];
      #pragma unroll
      for (int j = 0; j < 8; ++j) {
        Av[j].x = ekv_g(xv0 - w0[j]);
        Av[j].y = ekv_g(xv1 - w1[j]);
      }

      // --- 8 back-to-back WMMAs, independent accumulators: no hazard NOPs,
      //     exact fp32: D[m][n] = sum_k A[m][k]*1 + C[m][n] ---
      #pragma unroll
      for (int j = 0; j < 8; ++j) {
        acc[j] = __builtin_amdgcn_wmma_f32_16x16x4_f32(
            false, Av[j], false, Bones, (short)0, acc[j], false, false);
      }
    }

    // C/D layout: lane 0 holds D[M=0..7][N=0] in v0..v7; lane 16 holds M=8..15
    if (mrow == 0) {
      int prow = (pg << 4) + ((lane >> 4) << 3);
      #pragma unroll
      for (int j = 0; j < 8; ++j) {
        int co = cobase + j;
        float* op = out + ((n * K_COUT + co) * K_H + h) * K_W + prow;
        #pragma unroll
        for (int r = 0; r < 8; ++r) op[r] = acc[j][r] * SCALE;
      }
    }
  }
}

extern "C" void kernel_launch(void* const* d_in, const int* in_sizes, int n_in,
                              void* d_out, int out_size, void* d_ws, size_t ws_size,
                              hipStream_t stream) {
  (void)in_sizes; (void)n_in; (void)out_size; (void)d_ws; (void)ws_size;
  const float* x     = (const float*)d_in[0];
  const float* theta = (const float*)d_in[1];
  float* out         = (float*)d_out;

  const size_t shmem = (size_t)(K_THETA_ELTS + K_XS_ELTS) * sizeof(float); // 173,568 B
  hipFuncSetAttribute((const void*)ekv_conv_kernel,
                      hipFuncAttributeMaxDynamicSharedMemorySize, (int)shmem);
  ekv_conv_kernel<<<dim3(K_N * K_H), dim3(256), shmem, stream>>>(x, theta, out);
}